// BasesDecomposition_88716844466598
// MI455X (gfx1250) — compile-verified
//
#include <hip/hip_runtime.h>
#include <hip/hip_bf16.h>

typedef __attribute__((ext_vector_type(2))) float v2f;
typedef __attribute__((ext_vector_type(8))) float v8f;

#define DIM   128
#define NBASE 4

// ---------------------------------------------------------------------------
// Zero the output accumulator (harness poisons d_out with 0xAA).
// ---------------------------------------------------------------------------
__global__ void rgcn_zero(float* __restrict__ p, int n) {
    int i = blockIdx.x * blockDim.x + threadIdx.x;
    int stride = gridDim.x * blockDim.x;
    for (; i < n; i += stride) p[i] = 0.0f;
}

// ---------------------------------------------------------------------------
// Phase 1: y[n, b, o] = sum_i x[n, i] * bases[b, i, o]
// GEMM (N x 128) x (128 x 512) via V_WMMA_F32_16X16X4_F32 (exact f32).
// grid = (32 column tiles, CH), block = 128 (4 waves). Each wave owns one
// column tile (b, o0) and strides over row tiles, keeping all 32 B-fragments
// (K=128) resident in 64 VGPRs across row tiles.
// ---------------------------------------------------------------------------
__global__ void rgcn_xw_wmma(const float* __restrict__ x,
                             const float* __restrict__ bases,
                             float* __restrict__ y,
                             int n_tiles) {
    const int lane = threadIdx.x & 31;
    const int wave = threadIdx.x >> 5;
    const int col  = blockIdx.x;          // 0..31  -> (b, 16-col block)
    const int b    = col >> 3;
    const int o0   = (col & 7) << 4;
    const int half = lane >> 4;           // 0: lanes 0-15, 1: lanes 16-31
    const int l16  = lane & 15;

    const float* W = bases + (size_t)b * (DIM * DIM);

    // B fragment (4x16 f32 per k-step): lane half selects K pair {0,1}/{2,3}
    v2f bf[32];
#pragma unroll
    for (int ks = 0; ks < 32; ++ks) {
        const int k = ks * 4 + half * 2;
        bf[ks].x = W[(k + 0) * DIM + o0 + l16];
        bf[ks].y = W[(k + 1) * DIM + o0 + l16];
    }

    const int wstride = gridDim.y * 4;
    for (int t = blockIdx.y * 4 + wave; t < n_tiles; t += wstride) {
        // A fragment source: row = t*16 + (lane&15), K pair chosen by half
        const float* xr = x + ((size_t)t * 16 + l16) * DIM + half * 2;
        v8f acc = {};
#pragma unroll
        for (int ks = 0; ks < 32; ++ks) {
            v2f a;
            a.x = xr[ks * 4 + 0];
            a.y = xr[ks * 4 + 1];
            acc = __builtin_amdgcn_wmma_f32_16x16x4_f32(
                false, a, false, bf[ks], (short)0, acc, false, false);
        }
        // D layout: VGPR r, lane l -> M = r + 8*half, N = l16
        float* yp = y + ((size_t)t * 16 + half * 8) * (NBASE * DIM)
                      + b * DIM + o0 + l16;
#pragma unroll
        for (int r = 0; r < 8; ++r)
            yp[(size_t)r * (NBASE * DIM)] = acc[r];
    }
}

// ---------------------------------------------------------------------------
// Phase 2: per edge e: out[tgt] += ew[e] * sum_b bw[et,b] * y[src, b, :]
// One wave per edge; lane owns a float4 column slice. All traffic L2-resident.
// ---------------------------------------------------------------------------
__global__ void rgcn_edge_scatter(const float* __restrict__ y,
                                  const int* __restrict__ source,
                                  const int* __restrict__ target,
                                  const int* __restrict__ etype,
                                  const float* __restrict__ ew,
                                  const float* __restrict__ bw,  // (16,4)
                                  float* __restrict__ out,
                                  int num_edges) {
    const int lane   = threadIdx.x & 31;
    const int wave   = threadIdx.x >> 5;
    const int wpb    = blockDim.x >> 5;
    const int stride = gridDim.x * wpb;

    for (int e = blockIdx.x * wpb + wave; e < num_edges; e += stride) {
        const int   src = source[e];
        const int   tgt = target[e];
        const int   et  = etype[e];
        const float w   = ew[e];
        const float c0 = w * bw[et * 4 + 0];
        const float c1 = w * bw[et * 4 + 1];
        const float c2 = w * bw[et * 4 + 2];
        const float c3 = w * bw[et * 4 + 3];

        const float4* yr = reinterpret_cast<const float4*>(
            y + (size_t)src * (NBASE * DIM));
        const float4 v0 = yr[lane];           // b=0, o = lane*4..lane*4+3
        const float4 v1 = yr[32 + lane];      // b=1
        const float4 v2 = yr[64 + lane];      // b=2
        const float4 v3 = yr[96 + lane];      // b=3

        float4 a;
        a.x = c0 * v0.x + c1 * v1.x + c2 * v2.x + c3 * v3.x;
        a.y = c0 * v0.y + c1 * v1.y + c2 * v2.y + c3 * v3.y;
        a.z = c0 * v0.z + c1 * v1.z + c2 * v2.z + c3 * v3.z;
        a.w = c0 * v0.w + c1 * v1.w + c2 * v2.w + c3 * v3.w;

        float* op = out + (size_t)tgt * DIM + lane * 4;
        atomicAdd(op + 0, a.x);
        atomicAdd(op + 1, a.y);
        atomicAdd(op + 2, a.z);
        atomicAdd(op + 3, a.w);
    }
}

// ---------------------------------------------------------------------------
// Fallback (no workspace / N not tile-aligned): direct per-edge contraction.
// Correct but slow; taken if ws_size is insufficient or N is not a multiple of 16.
// ---------------------------------------------------------------------------
__global__ void rgcn_edge_direct(const float* __restrict__ x,
                                 const float* __restrict__ bases,
                                 const int* __restrict__ source,
                                 const int* __restrict__ target,
                                 const int* __restrict__ etype,
                                 const float* __restrict__ ew,
                                 const float* __restrict__ bw,
                                 float* __restrict__ out,
                                 int num_edges) {
    __shared__ float xs[DIM];
    const int o = threadIdx.x;  // blockDim.x == DIM
    for (int e = blockIdx.x; e < num_edges; e += gridDim.x) {
        const int   src = source[e];
        const int   tgt = target[e];
        const int   et  = etype[e];
        const float w   = ew[e];
        __syncthreads();
        xs[o] = x[(size_t)src * DIM + o];
        __syncthreads();
        float acc = 0.0f;
#pragma unroll
        for (int b = 0; b < NBASE; ++b) {
            const float* Wb = bases + (size_t)b * (DIM * DIM) + o;
            float t = 0.0f;
            for (int i = 0; i < DIM; ++i) t += xs[i] * Wb[(size_t)i * DIM];
            acc += w * bw[et * 4 + b] * t;
        }
        atomicAdd(&out[(size_t)tgt * DIM + o], acc);
    }
}

// ---------------------------------------------------------------------------
extern "C" void kernel_launch(void* const* d_in, const int* in_sizes, int n_in,
                              void* d_out, int out_size, void* d_ws, size_t ws_size,
                              hipStream_t stream) {
    const float* x      = (const float*)d_in[0];
    const int*   source = (const int*)  d_in[1];
    const int*   target = (const int*)  d_in[2];
    const int*   etype  = (const int*)  d_in[3];
    const float* ew     = (const float*)d_in[4];
    const float* bw     = (const float*)d_in[5];  // (R=16, B=4)
    const float* bases  = (const float*)d_in[6];  // (B=4, 128, 128)
    float* out = (float*)d_out;

    const int N = in_sizes[0] / DIM;   // 50000
    const int E = in_sizes[1];         // 1,600,000

    // 1) zero the output accumulator
    rgcn_zero<<<512, 256, 0, stream>>>(out, N * DIM);

    const size_t ybytes = (size_t)N * NBASE * DIM * sizeof(float);  // ~102 MB
    if (ws_size >= ybytes && (N % 16) == 0) {
        float* y = (float*)d_ws;
        // 2) y = x @ bases (stacked): WMMA f32 16x16x4
        const int n_tiles = N / 16;  // 3125
        rgcn_xw_wmma<<<dim3(32, 64), 128, 0, stream>>>(x, bases, y, n_tiles);
        // 3) edge gather/combine/scatter, one wave per edge (8 waves/block)
        const int wpb = 8;
        int blocks = (E + wpb - 1) / wpb;
        rgcn_edge_scatter<<<blocks, 32 * wpb, 0, stream>>>(
            y, source, target, etype, ew, bw, out, E);
    } else {
        rgcn_edge_direct<<<8192, DIM, 0, stream>>>(
            x, bases, source, target, etype, ew, bw, out, E);
    }
}